// DeepRouter_12060268167911
// MI455X (gfx1250) — compile-verified
//
#include <hip/hip_runtime.h>
#include <hip/hip_bf16.h>

typedef __attribute__((ext_vector_type(2))) float v2f;
typedef __attribute__((ext_vector_type(8))) float v8f;

#define TOKENS   16384
#define EXPERTS  64
#define DIM      4096
#define TOPK     8

// ---------------------------------------------------------------------------
// Stage 1: logits[t][e] = dot(x[t], W[e])   (bias folded in at stage 2)
// Block = 256 threads = 8 waves. Wave w owns row tile of 16 tokens x 64 experts.
// Per K-step of 4: 1 A-load (reused 4x) + 4 B-loads + 4 v_wmma_f32_16x16x4_f32.
// A 16x4 f32 layout: lane (half=l>>4, m=l&15) holds K = {2*half, 2*half+1} of
// row M=m  -> one float2 load from row-major x.
// B 4x16 f32 layout: lane (half,m) holds K = {2*half, 2*half+1} of col N=m,
// and B = W^T  -> one float2 load from row-major W.
// ---------------------------------------------------------------------------
__global__ __launch_bounds__(256) void moe_gemm(const float* __restrict__ x,
                                                const float* __restrict__ W,
                                                float* __restrict__ logits) {
  const int lane = threadIdx.x & 31;
  const int wave = threadIdx.x >> 5;
  const int rowTile = blockIdx.x * 8 + wave;
  const int row0 = rowTile * 16;
  const int half = lane >> 4;
  const int m = lane & 15;

  const float* __restrict__ xrow = x + (size_t)(row0 + m) * DIM + 2 * half;
  const float* __restrict__ wrow = W + (size_t)m * DIM + 2 * half;

  v8f acc0 = {}, acc1 = {}, acc2 = {}, acc3 = {};

#pragma unroll 8
  for (int kb = 0; kb < DIM; kb += 4) {
    v2f a  = *(const v2f*)(xrow + kb);
    v2f b0 = *(const v2f*)(wrow + kb);
    v2f b1 = *(const v2f*)(wrow + (size_t)16 * DIM + kb);
    v2f b2 = *(const v2f*)(wrow + (size_t)32 * DIM + kb);
    v2f b3 = *(const v2f*)(wrow + (size_t)48 * DIM + kb);
    acc0 = __builtin_amdgcn_wmma_f32_16x16x4_f32(false, a, false, b0, (short)0, acc0, false, false);
    acc1 = __builtin_amdgcn_wmma_f32_16x16x4_f32(false, a, false, b1, (short)0, acc1, false, false);
    acc2 = __builtin_amdgcn_wmma_f32_16x16x4_f32(false, a, false, b2, (short)0, acc2, false, false);
    acc3 = __builtin_amdgcn_wmma_f32_16x16x4_f32(false, a, false, b3, (short)0, acc3, false, false);
  }

  // C/D 16x16 f32 layout: VGPR g, lanes 0-15 -> token M=g, lanes 16-31 -> M=g+8,
  // N = m. Store logits row-major [TOKENS][EXPERTS].
  float* __restrict__ out = logits + (size_t)(row0 + 8 * half) * EXPERTS + m;
#pragma unroll
  for (int g = 0; g < 8; ++g) {
    out[(size_t)g * EXPERTS + 0]  = acc0[g];
    out[(size_t)g * EXPERTS + 16] = acc1[g];
    out[(size_t)g * EXPERTS + 32] = acc2[g];
    out[(size_t)g * EXPERTS + 48] = acc3[g];
  }
}

// ---------------------------------------------------------------------------
// Stage 2: softmax over 64 experts + top-8 per token (one wave per token).
// Each lane holds experts {lane, lane+32}. Writes indices (as float) to
// out_idx, raw top-8 softmax values to vals, per-block sum to partials.
// ---------------------------------------------------------------------------
__global__ __launch_bounds__(256) void moe_topk(const float* __restrict__ logits,
                                                const float* __restrict__ bias,
                                                float* __restrict__ out_idx,
                                                float* __restrict__ vals,
                                                float* __restrict__ partials) {
  const int lane  = threadIdx.x & 31;
  const int wave  = threadIdx.x >> 5;
  const int token = blockIdx.x * 8 + wave;

  float l0 = logits[(size_t)token * EXPERTS + lane]      + bias[lane];
  float l1 = logits[(size_t)token * EXPERTS + 32 + lane] + bias[lane + 32];

  // wave allreduce max (wave32)
  float mx = fmaxf(l0, l1);
#pragma unroll
  for (int off = 16; off; off >>= 1) mx = fmaxf(mx, __shfl_xor(mx, off));

  float e0 = __expf(l0 - mx);
  float e1 = __expf(l1 - mx);
  float s = e0 + e1;
#pragma unroll
  for (int off = 16; off; off >>= 1) s += __shfl_xor(s, off);
  float inv = 1.0f / s;
  float p0 = e0 * inv;
  float p1 = e1 * inv;

  float tsum = 0.0f;
#pragma unroll
  for (int j = 0; j < TOPK; ++j) {
    // local best (tie -> lower index; lane < lane+32 so p0 wins ties)
    float v = p0; int id = lane;
    if (p1 > v) { v = p1; id = lane + 32; }
    // wave allreduce argmax, tie-break on lower index
#pragma unroll
    for (int off = 16; off; off >>= 1) {
      float ov = __shfl_xor(v, off);
      int   oi = __shfl_xor(id, off);
      if (ov > v || (ov == v && oi < id)) { v = ov; id = oi; }
    }
    if (lane == 0) {
      out_idx[token * TOPK + j] = (float)id;  // exact for 0..63
      vals[token * TOPK + j]    = v;
    }
    tsum += v;  // same value in every lane
    // retire winner
    if (id == lane)      p0 = -1.0f;
    if (id == lane + 32) p1 = -1.0f;
  }

  __shared__ float ps[8];
  if (lane == 0) ps[wave] = tsum;
  __syncthreads();
  if (threadIdx.x == 0) {
    float t = 0.0f;
#pragma unroll
    for (int i = 0; i < 8; ++i) t += ps[i];
    partials[blockIdx.x] = t;
  }
}

// ---------------------------------------------------------------------------
// Stage 3: every block reduces the 2048 partials, then scales its chunk of
// vals by 1/globalsum into the second half of d_out. Grid: 256 x 256 threads,
// 2 elements per thread (131072 total).
// ---------------------------------------------------------------------------
__global__ __launch_bounds__(256) void moe_norm(const float* __restrict__ vals,
                                                const float* __restrict__ partials,
                                                float* __restrict__ out_w) {
  __shared__ float red[256];
  float s = 0.0f;
  for (int i = threadIdx.x; i < 2048; i += 256) s += partials[i];
  red[threadIdx.x] = s;
  __syncthreads();
#pragma unroll
  for (int off = 128; off; off >>= 1) {
    if (threadIdx.x < off) red[threadIdx.x] += red[threadIdx.x + off];
    __syncthreads();
  }
  const float invTotal = 1.0f / red[0];

  const int base = blockIdx.x * 512 + threadIdx.x;
  out_w[base]       = vals[base]       * invTotal;
  out_w[base + 256] = vals[base + 256] * invTotal;
}

extern "C" void kernel_launch(void* const* d_in, const int* in_sizes, int n_in,
                              void* d_out, int out_size, void* d_ws, size_t ws_size,
                              hipStream_t stream) {
  const float* x = (const float*)d_in[0];
  const float* W = (const float*)d_in[1];
  const float* b = (const float*)d_in[2];
  float* out = (float*)d_out;                    // [idx (T*K) | weights (T*K)]

  float* logits   = (float*)d_ws;                        // T*E floats = 4 MB
  float* vals     = logits + (size_t)TOKENS * EXPERTS;   // T*K floats = 512 KB
  float* partials = vals + (size_t)TOKENS * TOPK;        // 2048 floats

  (void)in_sizes; (void)n_in; (void)out_size; (void)ws_size;

  moe_gemm<<<TOKENS / (16 * 8), 256, 0, stream>>>(x, W, logits);
  moe_topk<<<TOKENS / 8, 256, 0, stream>>>(logits, b, out, vals, partials);
  moe_norm<<<256, 256, 0, stream>>>(vals, partials, out + (size_t)TOKENS * TOPK);
}